// MARN1_20504173871352
// MI455X (gfx1250) — compile-verified
//
#include <hip/hip_runtime.h>
#include <hip/hip_bf16.h>

// ---------------------------------------------------------------------------
// MARN forward for MI455X (gfx1250, wave32). GEMMs via V_WMMA_F32_16X16X4_F32,
// attention K/V staging via the Tensor Data Mover (tensor_load_to_lds).
// ---------------------------------------------------------------------------

typedef float v2f __attribute__((ext_vector_type(2)));
typedef float v8f __attribute__((ext_vector_type(8)));
typedef unsigned int v4u __attribute__((ext_vector_type(4)));
typedef int v4i __attribute__((ext_vector_type(4)));
typedef int v8i __attribute__((ext_vector_type(8)));

static __device__ inline v8f wmma4(v2f a, v2f b, v8f c) {
  // D = A(16x4,f32) * B(4x16,f32) + C(16x16,f32)
  return __builtin_amdgcn_wmma_f32_16x16x4_f32(false, a, false, b, (short)0, c,
                                               false, false);
}

// Issue one 2D-tile TDM load: global (tileH rows x tileW f32, row stride
// rowStride f32) -> LDS at ldsAddr (packed rows). D# layout per ISA 8.3/8.4.
// This toolchain exposes the 6-arg builtin: (g0, g1, g2, g3, g2b, cpol).
static __device__ inline void tdm_load_2d(unsigned ldsAddr, const void* gptr,
                                          unsigned tileW, unsigned tileH,
                                          unsigned rowStride) {
  unsigned long long ga = (unsigned long long)(size_t)gptr;
  v4u g0;
  g0.x = 0x1u;                                    // count=1, user mode
  g0.y = ldsAddr;                                 // lds_addr [63:32]
  g0.z = (unsigned)(ga & 0xffffffffu);            // global_addr low
  g0.w = (unsigned)((ga >> 32) & 0x1ffffffu) | 0x80000000u;  // addr hi | type=2
  v8i g1;
  g1[0] = (int)(2u << 16);                        // workgroup_mask=0, data_size=4B
  g1[1] = (int)((tileW & 0xffffu) << 16);         // tensor_dim0 low16 at [63:48]
  g1[2] = (int)((tileW >> 16) | ((tileH & 0xffffu) << 16));  // dim0 hi | dim1 lo
  g1[3] = (int)((tileH >> 16) | ((tileW & 0xffffu) << 16));  // dim1 hi | tile_dim0
  g1[4] = (int)(tileH & 0xffffu);                 // tile_dim1 (tile_dim2=0)
  g1[5] = (int)rowStride;                         // tensor_dim0_stride low32
  g1[6] = 0;                                      // stride hi | dim1_stride lo
  g1[7] = 0;
  v4i z4 = {0, 0, 0, 0};
  v8i z8 = {0, 0, 0, 0, 0, 0, 0, 0};
  __builtin_amdgcn_tensor_load_to_lds(g0, g1, z4, z4, z8, 0);
}

// ----------------------------- generic GEMM --------------------------------
// C[M,N] = act( opA(A)[M,K] @ B[K,N] + bias + (ACC ? C : 0) + res )
// flags: 1=ACC, 2=RELU, 4=RED gather on A's k index (power-of-2 group size):
//        a_col(k) = (k >> aShift) * aStride + aOff + (k & ((1<<aShift)-1))
#define GF_ACC  1
#define GF_RELU 2
#define GF_RED  4

__global__ __launch_bounds__(256) void gemm_kernel(
    const float* __restrict__ A, int lda,
    const float* __restrict__ B, int ldb,
    float* __restrict__ C, int ldc,
    const float* __restrict__ bias,
    const float* __restrict__ res, int ldres,
    int M, int N, int K, int flags, int aShift, int aStride, int aOff) {
  extern __shared__ float smem[];
  float* As = smem;             // [64][17]
  float* Bs = smem + 64 * 17;   // [16][65]

  const int t = threadIdx.x;
  const int lane = t & 31;
  const int wave = t >> 5;          // 0..7
  const int half = lane >> 4;       // 0/1
  const int r = lane & 15;
  const int mb = blockIdx.y * 64;
  const int nb = blockIdx.x * 64;
  const int m0 = (wave & 3) * 16;   // wave's row tile inside block
  const int n0 = (wave >> 2) * 32;  // wave's 32-col strip (2 wmma tiles)
  const int aMask = (1 << aShift) - 1;

  v8f acc0 = {};
  v8f acc1 = {};

  for (int k0 = 0; k0 < K; k0 += 16) {
#pragma unroll
    for (int i = 0; i < 4; ++i) {
      int id = t + i * 256;
      // A tile 64x16
      int ar = id >> 4, ac = id & 15;
      int gr = mb + ar, gk = k0 + ac;
      float va = 0.f;
      if (gr < M && gk < K) {
        long aix;
        if (flags & GF_RED)
          aix = (long)gr * lda + (long)(gk >> aShift) * aStride + aOff + (gk & aMask);
        else
          aix = (long)gr * lda + gk;
        va = A[aix];
      }
      As[ar * 17 + ac] = va;
      // B tile 16x64
      int br = id >> 6, bc = id & 63;
      int gkb = k0 + br, gc = nb + bc;
      float vb = 0.f;
      if (gkb < K && gc < N) {
        vb = B[(long)gkb * ldb + gc];
        if (gkb + 16 < K)  // prefetch next K tile -> global_prefetch_b8
          __builtin_prefetch(&B[(long)(gkb + 16) * ldb + gc], 0, 1);
      }
      Bs[br * 65 + bc] = vb;
    }
    __syncthreads();
#pragma unroll
    for (int kk = 0; kk < 4; ++kk) {
      v2f a, b0, b1;
      a.x = As[(m0 + r) * 17 + kk * 4 + 2 * half];
      a.y = As[(m0 + r) * 17 + kk * 4 + 2 * half + 1];
      b0.x = Bs[(kk * 4 + half) * 65 + n0 + r];
      b0.y = Bs[(kk * 4 + 2 + half) * 65 + n0 + r];
      b1.x = Bs[(kk * 4 + half) * 65 + n0 + 16 + r];
      b1.y = Bs[(kk * 4 + 2 + half) * 65 + n0 + 16 + r];
      acc0 = wmma4(a, b0, acc0);
      acc1 = wmma4(a, b1, acc1);
    }
    __syncthreads();
  }

#pragma unroll
  for (int v = 0; v < 8; ++v) {
    int row = mb + m0 + v + 8 * half;
    if (row >= M) continue;
    int col0 = nb + n0 + r;
    int col1 = col0 + 16;
    if (col0 < N) {
      float val = acc0[v];
      if (flags & GF_ACC) val += C[(long)row * ldc + col0];
      if (bias) val += bias[col0];
      if (res) val += res[(long)row * ldres + col0];
      if (flags & GF_RELU) val = fmaxf(val, 0.f);
      C[(long)row * ldc + col0] = val;
    }
    if (col1 < N) {
      float val = acc1[v];
      if (flags & GF_ACC) val += C[(long)row * ldc + col1];
      if (bias) val += bias[col1];
      if (res) val += res[(long)row * ldres + col1];
      if (flags & GF_RELU) val = fmaxf(val, 0.f);
      C[(long)row * ldc + col1] = val;
    }
  }
}

// ------------------------- fused attention (per b,h,rowblock) ---------------
// grid.x = B(128) * H(8) * RB(8 blocks of 32 rows). Q/K/V are [32768,320],
// head h occupies cols [h*40, h*40+40). S = softmax(QK^T/sqrt(40)); O = S V.
// K/V tiles (256x40) are staged into LDS by the Tensor Data Mover.
__global__ __launch_bounds__(256) void attn_kernel(
    const float* __restrict__ Q, const float* __restrict__ K,
    const float* __restrict__ V, float* __restrict__ O) {
  extern __shared__ float sm[];
  float* Ks = sm;                 // [256][40] (TDM packed rows)
  float* Vs = Ks + 256 * 40;      // [256][40]
  float* Qs = Vs + 256 * 40;      // [32][42]
  float* Ss = Qs + 32 * 42;       // [32][264]

  const int bid = blockIdx.x;
  const int rb = bid & 7;
  const int h = (bid >> 3) & 7;
  const int b = bid >> 6;
  const int t = threadIdx.x;
  const int lane = t & 31;
  const int wave = t >> 5;
  const int half = lane >> 4;
  const int r = lane & 15;
  const int rowBase = b * 256 + rb * 32;
  const int qc = h * 40;

  if (wave == 0) {  // TDM ignores EXEC; one wave issues both tile DMAs
    tdm_load_2d((unsigned)(size_t)Ks, K + (size_t)(b * 256) * 320 + qc, 40, 256, 320);
    tdm_load_2d((unsigned)(size_t)Vs, V + (size_t)(b * 256) * 320 + qc, 40, 256, 320);
  }
  for (int id = t; id < 32 * 40; id += 256) {
    int rr = id / 40, cc = id % 40;
    Qs[rr * 42 + cc] = Q[(long)(rowBase + rr) * 320 + qc + cc];
  }
  if (wave == 0) __builtin_amdgcn_s_wait_tensorcnt(0);
  __syncthreads();

  const float scale = 0.15811388300841898f;  // 1/sqrt(40)
  // S = Q @ K^T : 32x256 in tiles of 16x16 (2 x 16 tiles, 8 waves)
  for (int tile = wave; tile < 32; tile += 8) {
    int tr = tile >> 4, tc = tile & 15;
    int m0 = tr * 16, c0 = tc * 16;
    v8f acc = {};
#pragma unroll
    for (int kk = 0; kk < 10; ++kk) {  // K = 40
      v2f a, bf;
      a.x = Qs[(m0 + r) * 42 + kk * 4 + 2 * half];
      a.y = Qs[(m0 + r) * 42 + kk * 4 + 2 * half + 1];
      bf.x = Ks[(c0 + r) * 40 + kk * 4 + half];       // B[k][n] = K[n][k]
      bf.y = Ks[(c0 + r) * 40 + kk * 4 + 2 + half];
      acc = wmma4(a, bf, acc);
    }
#pragma unroll
    for (int v = 0; v < 8; ++v)
      Ss[(m0 + v + 8 * half) * 264 + c0 + r] = acc[v] * scale;
  }
  __syncthreads();

  if (t < 32) {  // row softmax over 256
    float* sr = Ss + t * 264;
    float mx = -1e30f;
    for (int c = 0; c < 256; ++c) mx = fmaxf(mx, sr[c]);
    float sum = 0.f;
    for (int c = 0; c < 256; ++c) { float e = __expf(sr[c] - mx); sr[c] = e; sum += e; }
    float inv = 1.f / sum;
    for (int c = 0; c < 256; ++c) sr[c] *= inv;
  }
  __syncthreads();

  // O = P @ V : 32x40 (2x3 tiles -> waves 0..5)
  if (wave < 6) {
    int tr = wave / 3, tc = wave % 3;
    int m0 = tr * 16, c0 = tc * 16;
    v8f acc = {};
    for (int kk = 0; kk < 64; ++kk) {  // K = 256
      v2f a, bf;
      a.x = Ss[(m0 + r) * 264 + kk * 4 + 2 * half];
      a.y = Ss[(m0 + r) * 264 + kk * 4 + 2 * half + 1];
      bf.x = Vs[(kk * 4 + half) * 40 + c0 + r];
      bf.y = Vs[(kk * 4 + 2 + half) * 40 + c0 + r];
      acc = wmma4(a, bf, acc);
    }
#pragma unroll
    for (int v = 0; v < 8; ++v) {
      int col = c0 + r;
      if (col < 40)
        O[(long)(rowBase + m0 + v + 8 * half) * 320 + qc + col] = acc[v];
    }
  }
}

// ------------------------------ layernorm ----------------------------------
__global__ __launch_bounds__(256) void ln_kernel(
    float* __restrict__ out, int ldo, const float* __restrict__ a, int ldA,
    const float* __restrict__ res, int ldr, const float* __restrict__ g,
    const float* __restrict__ bp, int D, int rows) {
  int row = blockIdx.x * 8 + (threadIdx.x >> 5);
  if (row >= rows) return;
  int lane = threadIdx.x & 31;
  float buf[16];
  float s = 0.f, s2 = 0.f;
  int i = 0;
  for (int c = lane; c < D; c += 32, ++i) {
    float v = a[(long)row * ldA + c] + (res ? res[(long)row * ldr + c] : 0.f);
    buf[i] = v; s += v; s2 += v * v;
  }
  for (int o = 16; o; o >>= 1) { s += __shfl_xor(s, o, 32); s2 += __shfl_xor(s2, o, 32); }
  float mean = s / D;
  float var = s2 / D - mean * mean;
  float rstd = rsqrtf(var + 1e-6f);
  i = 0;
  for (int c = lane; c < D; c += 32, ++i)
    out[(long)row * ldo + c] = (buf[i] - mean) * rstd * g[c] + bp[c];
}

// --------------------- scan elementwise helper kernels ----------------------
// fourdh is a power of two (64/256/512) -> pass its log2 to avoid div/mod.
__global__ void gather_sp_kernel(const float* __restrict__ qm, int* __restrict__ idx,
                                 const float* __restrict__ q, float* __restrict__ sp,
                                 int shift, int writeIdx) {
  int gid = blockIdx.x * blockDim.x + threadIdx.x;
  int fourdh = 1 << shift;
  if (gid >= 256 * fourdh) return;
  int row = gid >> shift, j = gid & (fourdh - 1);
  int id = qm[row * 2] >= qm[row * 2 + 1] ? 0 : 1;  // argmax, first on tie
  if (writeIdx && j == 0) idx[row] = id;
  sp[gid] = q[(row * 2 + id) * fourdh + j];
}

__global__ void gate_kernel(const float* __restrict__ s, float* __restrict__ c,
                            float* __restrict__ h, float* __restrict__ q,
                            const int* __restrict__ idx, float* __restrict__ cs,
                            int shift, int csOff) {
  int gid = blockIdx.x * blockDim.x + threadIdx.x;
  int fourdh = 1 << shift;
  int dh = fourdh >> 2;
  if (gid >= 256 * fourdh) return;
  int row = gid >> shift, j = gid & (fourdh - 1);
  float sv = s[row * fourdh + j];
  q[(row * 2 + idx[row]) * fourdh + j] = sv;  // q_mem[row, idx] = s[row]
  if (j < dh) {
    float f = 1.f / (1.f + __expf(-sv));
    float ii = 1.f / (1.f + __expf(-s[row * fourdh + dh + j]));
    float o = 1.f / (1.f + __expf(-s[row * fourdh + 2 * dh + j]));
    float ch = tanhf(s[row * fourdh + 3 * dh + j]);
    float cn = f * c[row * dh + j] + ii * ch;
    c[row * dh + j] = cn;
    h[row * dh + j] = tanhf(cn) * o;
    cs[row * 208 + csOff + j] = cn;
  }
}

__global__ void attsm_kernel(const float* __restrict__ logits,
                             const float* __restrict__ cs,
                             float* __restrict__ attended) {
  int wid = blockIdx.x * 8 + (threadIdx.x >> 5);  // (n,g) pair, 1024 total
  if (wid >= 1024) return;
  int lane = threadIdx.x & 31;
  int n = wid >> 2, g = wid & 3;
  const float* l = logits + n * 832 + g * 208;
  float mx = -1e30f;
  for (int c = lane; c < 208; c += 32) mx = fmaxf(mx, l[c]);
  for (int o = 16; o; o >>= 1) mx = fmaxf(mx, __shfl_xor(mx, o, 32));
  float sum = 0.f;
  for (int c = lane; c < 208; c += 32) sum += __expf(l[c] - mx);
  for (int o = 16; o; o >>= 1) sum += __shfl_xor(sum, o, 32);
  float inv = 1.f / sum;
  const float* csr = cs + n * 208;
  float* outp = attended + n * 832 + g * 208;
  for (int c = lane; c < 208; c += 32) outp[c] = __expf(l[c] - mx) * inv * csr[c];
}

__global__ void z_kernel(const float* __restrict__ cs, const float* __restrict__ tx,
                         float* __restrict__ z) {
  int wid = blockIdx.x * 8 + (threadIdx.x >> 5);  // row n
  if (wid >= 256) return;
  int lane = threadIdx.x & 31;
  const float* c = cs + wid * 208;
  const float* t = tx + wid * 208;
  float mx = -1e30f;
  for (int i = lane; i < 208; i += 32) mx = fmaxf(mx, c[i] * t[i]);
  for (int o = 16; o; o >>= 1) mx = fmaxf(mx, __shfl_xor(mx, o, 32));
  float sum = 0.f;
  for (int i = lane; i < 208; i += 32) sum += __expf(c[i] * t[i] - mx);
  for (int o = 16; o; o >>= 1) sum += __shfl_xor(sum, o, 32);
  float inv = 1.f / sum;  // alpha/sum(alpha) with sum==1 folds into inv
  for (int i = lane; i < 208; i += 32)
    z[wid * 208 + i] = __expf(c[i] * t[i] - mx) * inv * t[i];
}

__global__ void feats_kernel(const float* __restrict__ hl, const float* __restrict__ ha,
                             const float* __restrict__ hv, const float* __restrict__ z,
                             float* __restrict__ feats) {
  int gid = blockIdx.x * blockDim.x + threadIdx.x;
  if (gid >= 256 * 416) return;
  int row = gid / 416, j = gid % 416;
  float v;
  if (j < 128) v = hl[row * 128 + j];
  else if (j < 144) v = ha[row * 16 + (j - 128)];
  else if (j < 208) v = hv[row * 64 + (j - 144)];
  else v = z[row * 208 + (j - 208)];
  feats[gid] = v;
}

// ------------------------------- host side ----------------------------------
static const int GEMM_SMEM = (64 * 17 + 16 * 65) * 4;
static const size_t ATT_SMEM = (size_t)(256 * 40 + 256 * 40 + 32 * 42 + 32 * 264) * 4;

static void gemm(hipStream_t st, const float* A, int lda, const float* B, int ldb,
                 float* C, int ldc, const float* bias, const float* res, int ldres,
                 int M, int N, int K, int flags,
                 int aShift = 0, int aStride = 0, int aOff = 0) {
  dim3 g((N + 63) / 64, (M + 63) / 64);
  gemm_kernel<<<g, 256, GEMM_SMEM, st>>>(A, lda, B, ldb, C, ldc, bias, res, ldres,
                                         M, N, K, flags, aShift, aStride, aOff);
}

struct CellP { const float *Sb, *Sw, *Ub, *Uw, *Vb, *Vw, *Wb, *Ww; };
struct EncP  { const float *fcb, *fcw, *ln1b, *ln1g, *ln2b, *ln2g,
                           *w1b, *w1w, *w2b, *w2w, *wkb, *wkw, *wqb, *wqw, *wvb, *wvw; };

extern "C" void kernel_launch(void* const* d_in, const int* in_sizes, int n_in,
                              void* d_out, int out_size, void* d_ws, size_t ws_size,
                              hipStream_t stream) {
  (void)in_sizes; (void)n_in; (void)out_size; (void)ws_size;
  auto f = [&](int i) { return (const float*)d_in[i]; };

  // ---- input layout: x, qmask, then params leaves (dict keys sorted) ----
  const float* X = f(0);        // (128,256,712)
  const float* QM = f(1);       // (128,256,2)
  const float* att_b = f(2);  const float* att_w = f(3);      // (832),(208,832)
  CellP ca{f(4), f(5), f(6), f(7), f(8), f(9), f(10), f(11)};     // dh=16,din=100
  CellP cl{f(12), f(13), f(14), f(15), f(16), f(17), f(18), f(19)};// dh=128,din=100
  CellP cv{f(20), f(21), f(22), f(23), f(24), f(25), f(26), f(27)};// dh=64,din=512
  EncP ea{f(28), f(29), f(30), f(31), f(32), f(33), f(34), f(35),
          f(36), f(37), f(38), f(39), f(40), f(41), f(42), f(43)}; // d=100
  EncP el{f(44), f(45), f(46), f(47), f(48), f(49), f(50), f(51),
          f(52), f(53), f(54), f(55), f(56), f(57), f(58), f(59)}; // d=100
  EncP ev{f(60), f(61), f(62), f(63), f(64), f(65), f(66), f(67),
          f(68), f(69), f(70), f(71), f(72), f(73), f(74), f(75)}; // d=512
  const float *ga1b = f(76), *ga1w = f(77);   // (64),(244,64)
  const float *ga2b = f(78), *ga2w = f(79);   // (208),(64,208)
  const float *o1b = f(80), *o1w = f(81);     // (32),(416,32)
  const float *o2b = f(82), *o2w = f(83);     // (6),(32,6)
  const float *rab = f(84), *raw = f(85);     // (128),(64,128)
  const float *rlb = f(86), *rlw = f(87);     // (16),(512,16)
  const float *rvb = f(88), *rvw = f(89);     // (100),(256,100)

  // ---- workspace bump allocator ----
  char* wsp = (char*)d_ws;
  auto alloc = [&](size_t nf) {
    float* p = (float*)wsp;
    wsp += ((nf * 4 + 255) / 256) * 256;
    return p;
  };
  const int TN = 128 * 256;  // 32768 rows
  float* xe_l = alloc((size_t)TN * 100);
  float* xe_a = alloc((size_t)TN * 100);
  float* xe_v = alloc((size_t)TN * 512);
  float* bQ = alloc((size_t)TN * 320);
  float* bK = alloc((size_t)TN * 320);
  float* bV = alloc((size_t)TN * 320);
  float* bO = alloc((size_t)TN * 320);
  float* t1 = alloc((size_t)TN * 512);
  float* hid = alloc((size_t)TN * 50);
  float* s_l = alloc(256 * 512); float* s_a = alloc(256 * 64); float* s_v = alloc(256 * 256);
  float* c_l = alloc(256 * 128); float* h_l = alloc(256 * 128);
  float* c_a = alloc(256 * 16);  float* h_a = alloc(256 * 16);
  float* c_v = alloc(256 * 64);  float* h_v = alloc(256 * 64);
  float* zb = alloc(256 * 208);
  float* q_l = alloc(256 * 2 * 512); float* q_a = alloc(256 * 2 * 64); float* q_v = alloc(256 * 2 * 256);
  float* sp_l = alloc(256 * 512); float* sp_a = alloc(256 * 64); float* sp_v = alloc(256 * 256);
  float* cs = alloc(256 * 208);
  float* attlog = alloc(256 * 832);
  float* attd = alloc(256 * 832);
  float* red = alloc(256 * 244);
  float* gah = alloc(256 * 64);
  float* tx = alloc(256 * 208);
  float* feats = alloc(256 * 416);
  float* o1 = alloc(256 * 32);
  int* idx = (int*)alloc(256);
  float* out = (float*)d_out;  // (128*256, 6)

  // ---- encoders (per modality) ----
  struct Mod { const EncP* p; int xoff; int D; float* xe; };
  Mod mods[3] = {{&el, 0, 100, xe_l}, {&ea, 100, 100, xe_a}, {&ev, 200, 512, xe_v}};
  for (int m = 0; m < 3; ++m) {
    const EncP& e = *mods[m].p;
    const float* xs = X + mods[m].xoff;  // lda 712
    int D = mods[m].D;
    float* xe = mods[m].xe;
    gemm(stream, xs, 712, e.wqw, 320, bQ, 320, e.wqb, nullptr, 0, TN, 320, D, 0);
    gemm(stream, xs, 712, e.wkw, 320, bK, 320, e.wkb, nullptr, 0, TN, 320, D, 0);
    gemm(stream, xs, 712, e.wvw, 320, bV, 320, e.wvb, nullptr, 0, TN, 320, D, 0);
    attn_kernel<<<128 * 8 * 8, 256, ATT_SMEM, stream>>>(bQ, bK, bV, bO);
    gemm(stream, bO, 320, e.fcw, D, t1, D, e.fcb, nullptr, 0, TN, D, 320, 0);
    ln_kernel<<<TN / 8, 256, 0, stream>>>(xe, D, t1, D, xs, 712, e.ln1g, e.ln1b, D, TN);
    gemm(stream, xe, D, e.w1w, 50, hid, 50, e.w1b, nullptr, 0, TN, 50, D, GF_RELU);
    gemm(stream, hid, 50, e.w2w, D, t1, D, e.w2b, nullptr, 0, TN, D, 50, 0);
    ln_kernel<<<TN / 8, 256, 0, stream>>>(xe, D, t1, D, xe, D, e.ln2g, e.ln2b, D, TN);
  }

  // ---- zero scan state ----
  (void)hipMemsetAsync(c_l, 0, 256 * 128 * 4, stream);
  (void)hipMemsetAsync(h_l, 0, 256 * 128 * 4, stream);
  (void)hipMemsetAsync(c_a, 0, 256 * 16 * 4, stream);
  (void)hipMemsetAsync(h_a, 0, 256 * 16 * 4, stream);
  (void)hipMemsetAsync(c_v, 0, 256 * 64 * 4, stream);
  (void)hipMemsetAsync(h_v, 0, 256 * 64 * 4, stream);
  (void)hipMemsetAsync(zb, 0, 256 * 208 * 4, stream);
  (void)hipMemsetAsync(q_l, 0, (size_t)256 * 2 * 512 * 4, stream);
  (void)hipMemsetAsync(q_a, 0, (size_t)256 * 2 * 64 * 4, stream);
  (void)hipMemsetAsync(q_v, 0, (size_t)256 * 2 * 256 * 4, stream);

  // ---- sequential scan over T=128 steps ----
  for (int t = 0; t < 128; ++t) {
    const float* qm_t = QM + (size_t)t * 256 * 2;
    gather_sp_kernel<<<(256 * 512) / 256, 256, 0, stream>>>(qm_t, idx, q_l, sp_l, 9, 1);
    gather_sp_kernel<<<(256 * 64) / 256, 256, 0, stream>>>(qm_t, idx, q_a, sp_a, 6, 0);
    gather_sp_kernel<<<(256 * 256) / 256, 256, 0, stream>>>(qm_t, idx, q_v, sp_v, 8, 0);

    const float* xl = xe_l + (size_t)t * 256 * 100;
    const float* xa = xe_a + (size_t)t * 256 * 100;
    const float* xv = xe_v + (size_t)t * 256 * 512;
    // s = xW + hU + zV + spS  (each linear carries its own bias)
    gemm(stream, xl, 100, cl.Ww, 512, s_l, 512, cl.Wb, nullptr, 0, 256, 512, 100, 0);
    gemm(stream, h_l, 128, cl.Uw, 512, s_l, 512, cl.Ub, nullptr, 0, 256, 512, 128, GF_ACC);
    gemm(stream, zb, 208, cl.Vw, 512, s_l, 512, cl.Vb, nullptr, 0, 256, 512, 208, GF_ACC);
    gemm(stream, sp_l, 512, cl.Sw, 512, s_l, 512, cl.Sb, nullptr, 0, 256, 512, 512, GF_ACC);

    gemm(stream, xa, 100, ca.Ww, 64, s_a, 64, ca.Wb, nullptr, 0, 256, 64, 100, 0);
    gemm(stream, h_a, 16, ca.Uw, 64, s_a, 64, ca.Ub, nullptr, 0, 256, 64, 16, GF_ACC);
    gemm(stream, zb, 208, ca.Vw, 64, s_a, 64, ca.Vb, nullptr, 0, 256, 64, 208, GF_ACC);
    gemm(stream, sp_a, 64, ca.Sw, 64, s_a, 64, ca.Sb, nullptr, 0, 256, 64, 64, GF_ACC);

    gemm(stream, xv, 512, cv.Ww, 256, s_v, 256, cv.Wb, nullptr, 0, 256, 256, 512, 0);
    gemm(stream, h_v, 64, cv.Uw, 256, s_v, 256, cv.Ub, nullptr, 0, 256, 256, 64, GF_ACC);
    gemm(stream, zb, 208, cv.Vw, 256, s_v, 256, cv.Vb, nullptr, 0, 256, 256, 208, GF_ACC);
    gemm(stream, sp_v, 256, cv.Sw, 256, s_v, 256, cv.Sb, nullptr, 0, 256, 256, 256, GF_ACC);

    gate_kernel<<<(256 * 512) / 256, 256, 0, stream>>>(s_l, c_l, h_l, q_l, idx, cs, 9, 0);
    gate_kernel<<<(256 * 64) / 256, 256, 0, stream>>>(s_a, c_a, h_a, q_a, idx, cs, 6, 128);
    gate_kernel<<<(256 * 256) / 256, 256, 0, stream>>>(s_v, c_v, h_v, q_v, idx, cs, 8, 144);

    // attention over concatenated cell states
    gemm(stream, cs, 208, att_w, 832, attlog, 832, att_b, nullptr, 0, 256, 832, 208, 0);
    attsm_kernel<<<128, 256, 0, stream>>>(attlog, cs, attd);
    // reductions read attended[:, :, slice].reshape(n,-1) via RED gather mode
    gemm(stream, attd, 832, rlw, 16, red + 0, 244, rlb, nullptr, 0, 256, 16, 512,
         GF_RED, 7, 208, 0);
    gemm(stream, attd, 832, raw, 128, red + 16, 244, rab, nullptr, 0, 256, 128, 64,
         GF_RED, 4, 208, 128);
    gemm(stream, attd, 832, rvw, 100, red + 144, 244, rvb, nullptr, 0, 256, 100, 256,
         GF_RED, 6, 208, 144);

    gemm(stream, red, 244, ga1w, 64, gah, 64, ga1b, nullptr, 0, 256, 64, 244, GF_RELU);
    gemm(stream, gah, 64, ga2w, 208, tx, 208, ga2b, nullptr, 0, 256, 208, 64, 0);
    z_kernel<<<32, 256, 0, stream>>>(cs, tx, zb);

    feats_kernel<<<(256 * 416 + 255) / 256, 256, 0, stream>>>(h_l, h_a, h_v, zb, feats);
    gemm(stream, feats, 416, o1w, 32, o1, 32, o1b, nullptr, 0, 256, 32, 416, GF_RELU);
    gemm(stream, o1, 32, o2w, 6, out + (size_t)t * 256 * 6, 6, o2b, nullptr, 0,
         256, 6, 32, 0);
  }
}